// MMNet_base_71451075936654
// MI455X (gfx1250) — compile-verified
//
#include <hip/hip_runtime.h>

typedef float v2f __attribute__((ext_vector_type(2)));
typedef float v4f __attribute__((ext_vector_type(4)));
typedef float v8f __attribute__((ext_vector_type(8)));

#define NR2 128
#define NT2 64
#define MCON 8

__device__ __forceinline__ float wave_reduce_sum(float v) {
    #pragma unroll
    for (int off = 16; off > 0; off >>= 1)
        v += __shfl_xor(v, off, 32);
    return v;
}

__global__ __launch_bounds__(256) void mmnet_wmma_kernel(
    const float* __restrict__ H,             // [B,128,64]
    const float* __restrict__ y,             // [B,128]
    const float* __restrict__ x_out,         // [B,64]
    const float* __restrict__ noise_sigma,   // [B]
    const float* __restrict__ constellation, // [8]
    const float* __restrict__ theta1,        // [1]
    const float* __restrict__ gamma,         // [64]
    float* __restrict__ out)                 // [B,64]
{
    __shared__ float sH[NR2 * NT2];   // 32 KB, row-major H
    __shared__ float sX[NT2];
    __shared__ float sRT[NR2];
    __shared__ float sZT[NT2];
    __shared__ float sRedH[8];
    __shared__ float sRedR[4];
    __shared__ float sRedG[8];
    __shared__ float sTau[1];

    const int b    = blockIdx.x;
    const int tid  = threadIdx.x;
    const int lane = tid & 31;
    const int wave = tid >> 5;

    const float* Hb = H + (size_t)b * (NR2 * NT2);

    // ---------- Phase 0: H -> LDS (non-temporal b128), fused sum(H^2) ----------
    float h2 = 0.f;
    {
        const v4f* Hb4 = (const v4f*)Hb;
        v4f* sH4 = (v4f*)sH;
        #pragma unroll
        for (int i = 0; i < 8; ++i) {
            int idx = tid + i * 256;                 // 2048 v4f total
            v4f v = __builtin_nontemporal_load(&Hb4[idx]);
            sH4[idx] = v;
            h2 += v.x * v.x + v.y * v.y + v.z * v.z + v.w * v.w;
        }
    }
    if (tid < NT2) sX[tid] = x_out[(size_t)b * NT2 + tid];
    h2 = wave_reduce_sum(h2);
    if (lane == 0) sRedH[wave] = h2;
    __syncthreads();

    // ---------- Phase 1: rt = y - H x, fused ||rt||^2 ----------
    float r2 = 0.f;
    if (tid < NR2) {
        float acc = 0.f;
        const float* hr = sH + tid * NT2;
        #pragma unroll 8
        for (int t = 0; t < NT2; ++t) acc += hr[t] * sX[t];
        float rt = y[(size_t)b * NR2 + tid] - acc;
        sRT[tid] = rt;
        r2 = rt * rt;
    }
    r2 = wave_reduce_sum(r2);
    if (lane == 0 && wave < 4) sRedR[wave] = r2;
    __syncthreads();

    // ---------- Phase 2: zt = x + theta1 * H^T rt (waves 0-1; EXEC uniform) ----------
    if (tid < NT2) {
        float acc = 0.f;
        #pragma unroll 8
        for (int r = 0; r < NR2; ++r) acc += sH[r * NT2 + tid] * sRT[r];
        sZT[tid] = sX[tid] + theta1[0] * acc;
    }

    // ---------- Phase 3: ||H^T H||_F^2 via V_WMMA_F32_16X16X4_F32 ----------
    // Wave w owns G-tiles (mi, ni0) and (mi, ni0+1); both share the A tile.
    // A(m,k) = H[k0+k][mi*16+m], B(k,n) = H[k0+k][n0+n]:
    //   lanes 0-15: K = {0,1} in {v0,v1}; lanes 16-31: K = {2,3}.
    {
        const int mi    = wave >> 1;
        const int ni0   = (wave & 1) * 2;
        const int colA  = mi * 16 + (lane & 15);
        const int colB0 = ni0 * 16 + (lane & 15);
        const int colB1 = colB0 + 16;
        const int krow  = (lane >> 4) * 2;

        v8f acc0 = {};
        v8f acc1 = {};
        #pragma unroll 4
        for (int k0 = 0; k0 < NR2; k0 += 4) {
            int base0 = (k0 + krow) * NT2;
            int base1 = base0 + NT2;
            v2f a, b0v, b1v;
            a.x   = sH[base0 + colA];   a.y   = sH[base1 + colA];
            b0v.x = sH[base0 + colB0];  b0v.y = sH[base1 + colB0];
            b1v.x = sH[base0 + colB1];  b1v.y = sH[base1 + colB1];
            acc0 = __builtin_amdgcn_wmma_f32_16x16x4_f32(
                false, a, false, b0v, (short)0, acc0, false, false);
            acc1 = __builtin_amdgcn_wmma_f32_16x16x4_f32(
                false, a, false, b1v, (short)0, acc1, false, false);
        }
        float g2 = 0.f;
        #pragma unroll
        for (int i = 0; i < 8; ++i) g2 += acc0[i] * acc0[i] + acc1[i] * acc1[i];
        g2 = wave_reduce_sum(g2);
        if (lane == 0) sRedG[wave] = g2;
    }
    __syncthreads();

    // ---------- Phase 4: scalars -> tau2 base ----------
    if (tid == 0) {
        float th = 0.f, rr = 0.f, gg = 0.f;
        #pragma unroll
        for (int i = 0; i < 8; ++i) th += sRedH[i];
        #pragma unroll
        for (int i = 0; i < 4; ++i) rr += sRedR[i];
        #pragma unroll
        for (int i = 0; i < 8; ++i) gg += sRedG[i];
        float sig  = noise_sigma[b];
        float sig2 = sig * sig;
        float v2 = (rr - (float)NR2 * sig2 * 0.5f) / th;
        v2 = fmaxf(v2, 1e-9f);
        float trCCT = gg - 2.f * th + (float)NT2;
        sTau[0] = (trCCT / (float)NT2) * v2 + sig2 / (2.f * (float)NT2) * th;
    }
    __syncthreads();

    // ---------- Phase 5: gaussian soft-VQ denoiser ----------
    if (tid < NT2) {
        float tau2 = sTau[0] / gamma[tid];
        float z    = sZT[tid];
        float inv  = -1.f / (2.f * tau2);
        float args[MCON];
        float amax = -1e30f;
        #pragma unroll
        for (int m = 0; m < MCON; ++m) {
            float c = constellation[m];
            float d = z - c;
            args[m] = d * d * inv;
            amax = fmaxf(amax, args[m]);
        }
        float se = 0.f, sec = 0.f;
        #pragma unroll
        for (int m = 0; m < MCON; ++m) {
            float e = __expf(args[m] - amax);
            se  += e;
            sec += e * constellation[m];
        }
        __builtin_nontemporal_store(sec / se, &out[(size_t)b * NT2 + tid]);
    }
}

extern "C" void kernel_launch(void* const* d_in, const int* in_sizes, int n_in,
                              void* d_out, int out_size, void* d_ws, size_t ws_size,
                              hipStream_t stream) {
    const float* H     = (const float*)d_in[0];
    const float* y     = (const float*)d_in[1];
    const float* x_out = (const float*)d_in[2];
    const float* sigma = (const float*)d_in[3];
    const float* cons  = (const float*)d_in[4];
    const float* th1   = (const float*)d_in[5];
    const float* gamma = (const float*)d_in[6];
    float* out = (float*)d_out;

    const int B = in_sizes[3];  // noise_sigma has one element per batch item

    mmnet_wmma_kernel<<<B, 256, 0, stream>>>(H, y, x_out, sigma, cons, th1, gamma, out);
}